// YoloV1Loss_79748952752501
// MI455X (gfx1250) — compile-verified
//
#include <hip/hip_runtime.h>
#include <stdint.h>

// ---------------------------------------------------------------------------
// YOLOv1 loss, gfx1250 (MI455X). Memory-bound: 96.3MB of fp32 activations,
// ~1.7 flop/byte -> HBM roofline ~4.1us @ 23.3TB/s. Strategy:
//   * CDNA5 async tensor path: GLOBAL_LOAD_ASYNC_TO_LDS_B128 stages each
//     block's contiguous 30,720B slab (256 cells x 120B) into LDS with
//     perfectly coalesced 16B/lane transfers (ASYNCcnt-tracked), then each
//     thread reads its 30-float AoS record from LDS (conflict-free).
//   * Deterministic two-pass reduction (block tree -> d_ws partials ->
//     single-block fixed-order sum). No float atomics.
// ---------------------------------------------------------------------------

#define SS        49          // S*S
#define NBOX      2
#define NCLS      20
#define CELL_F    30          // floats per cell (B*5 + C)
#define TPB       256
#define CELL_BYTES (CELL_F * 4)                  // 120
#define BLOCK_BYTES (TPB * CELL_BYTES)           // 30720
#define BLOCK_CHUNKS (BLOCK_BYTES / 16)          // 1920 x 16B lane-transfers
#define LAMBDA_COORD 5.0f
#define LAMBDA_NOOBJ 0.5f
#define EPSF 1e-10f

__device__ __forceinline__ float iou_f(const float* p, const float* g) {
    float ix1 = fmaxf(p[0], g[0]);
    float iy1 = fmaxf(p[1], g[1]);
    float ix2 = fminf(p[2], g[2]);
    float iy2 = fminf(p[3], g[3]);
    float inter = fmaxf(ix2 - ix1, 0.0f) * fmaxf(iy2 - iy1, 0.0f);
    float ap = (p[2] - p[0]) * (p[3] - p[1]);
    float ag = (g[2] - g[0]) * (g[3] - g[1]);
    return inter / fmaxf(ap + ag - inter, EPSF);
}

__global__ __launch_bounds__(TPB) void yolo_pass1(
    const float* __restrict__ outputs,
    const float* __restrict__ gt_boxes,
    const int*   __restrict__ gt_labels,
    float* __restrict__ partials)
{
    __shared__ __align__(16) float stage[TPB * CELL_F];  // 30720 B
    __shared__ float red[TPB];

    const int tid    = threadIdx.x;
    const int block0 = blockIdx.x * TPB;                 // first cell of block

    // ---- Stage: global -> LDS via CDNA5 async tensor path (16B per lane) ---
    {
        // Low 32 bits of the generic shared-pointer are the wave-relative LDS
        // byte address (aperture mapping), which is what VDST expects.
        uint32_t lds_base   = (uint32_t)(uintptr_t)&stage[0];
        uint32_t gbyte_base = (uint32_t)block0 * (uint32_t)CELL_BYTES;
        uint64_t gaddr      = (uint64_t)(uintptr_t)outputs;
        #pragma unroll
        for (int k = 0; k < 8; ++k) {
            int chunk = tid + k * TPB;
            if (chunk < BLOCK_CHUNKS) {                  // 1920 = 7.5 * 256
                uint32_t off = (uint32_t)chunk * 16u;
                asm volatile(
                    "global_load_async_to_lds_b128 %0, %1, %2"
                    :
                    : "v"(lds_base + off), "v"(gbyte_base + off), "s"(gaddr)
                    : "memory");
            }
        }
        asm volatile("s_wait_asynccnt 0" ::: "memory");  // this wave's DMAs done
    }
    __syncthreads();                                     // all waves' DMAs done

    // ---- Per-cell loss ----------------------------------------------------
    const float* my = &stage[tid * CELL_F];
    float pb[NBOX][5];
    #pragma unroll
    for (int i = 0; i < NBOX; ++i)
        #pragma unroll
        for (int j = 0; j < 5; ++j) pb[i][j] = my[i * 5 + j];
    float pc[NCLS];
    #pragma unroll
    for (int j = 0; j < NCLS; ++j) pc[j] = my[NBOX * 5 + j];

    const int cell = block0 + tid;
    const int b    = cell / SS;

    float g[NBOX][4];
    {
        const float* gb = gt_boxes + (size_t)b * (NBOX * 4);
        #pragma unroll
        for (int i = 0; i < NBOX; ++i)
            #pragma unroll
            for (int j = 0; j < 4; ++j) g[i][j] = gb[i * 4 + j];
    }
    const int lab0 = gt_labels[b * 2 + 0];
    const int lab1 = gt_labels[b * 2 + 1];

    // IoU of each pred box vs each GT; max over pred axis, argmax over GT axis
    float pg0 = fmaxf(iou_f(pb[0], g[0]), iou_f(pb[1], g[0]));
    float pg1 = fmaxf(iou_f(pb[0], g[1]), iou_f(pb[1], g[1]));
    float best_iou = fmaxf(pg0, pg1);
    int   bi = (pg1 > pg0) ? 1 : 0;      // argmax: first index wins ties
    bool  has_obj = best_iou > 0.0f;

    // Reference quirk: bi (a GT index) also selects the pred box (axis=2).
    float sp0 = pb[bi][0], sp1 = pb[bi][1], sp2 = pb[bi][2], sp3 = pb[bi][3];
    float spc = pb[bi][4];
    float sg0 = g[bi][0], sg1 = g[bi][1], sg2 = g[bi][2], sg3 = g[bi][3];

    float dx = sp0 - sg0, dy = sp1 - sg1;
    float dw = sqrtf(sp2) - sqrtf(sg2);
    float dh = sqrtf(sp3) - sqrtf(sg3);
    float loc = LAMBDA_COORD * (dx * dx + dy * dy + dw * dw + dh * dh);
    float dc  = spc - best_iou;
    float conf_obj = dc * dc;

    // log-softmax over 20 classes (max-subtracted, like jax.nn.log_softmax)
    float m = pc[0];
    #pragma unroll
    for (int j = 1; j < NCLS; ++j) m = fmaxf(m, pc[j]);
    float esum = 0.0f;
    #pragma unroll
    for (int j = 0; j < NCLS; ++j) esum += expf(pc[j] - m);
    float lse = m + logf(esum);
    // one-hot via .set(): duplicate labels still contribute a single 1.0
    float picked = 0.0f;
    #pragma unroll
    for (int j = 0; j < NCLS; ++j) {
        float sel = ((j == lab0) || (j == lab1)) ? 1.0f : 0.0f;
        picked += sel * pc[j];
    }
    float n_ones = (lab0 == lab1) ? 1.0f : 2.0f;
    float cls = -(picked - n_ones * lse);

    float noobj = LAMBDA_NOOBJ * (pb[0][4] * pb[0][4] + pb[1][4] * pb[1][4]);
    float loss = has_obj ? (loc + conf_obj + cls) : noobj;

    // ---- Deterministic block reduction ------------------------------------
    red[tid] = loss;
    __syncthreads();
    #pragma unroll
    for (int s = TPB / 2; s > 0; s >>= 1) {
        if (tid < s) red[tid] += red[tid + s];
        __syncthreads();
    }
    if (tid == 0) partials[blockIdx.x] = red[0];
}

__global__ __launch_bounds__(TPB) void yolo_pass2(
    const float* __restrict__ partials, float* __restrict__ out,
    int n, float inv_bt)
{
    __shared__ float red[TPB];
    float s = 0.0f;
    for (int i = threadIdx.x; i < n; i += TPB) s += partials[i];  // fixed order
    red[threadIdx.x] = s;
    __syncthreads();
    #pragma unroll
    for (int k = TPB / 2; k > 0; k >>= 1) {
        if (threadIdx.x < k) red[threadIdx.x] += red[threadIdx.x + k];
        __syncthreads();
    }
    if (threadIdx.x == 0) out[0] = red[0] * inv_bt;
}

extern "C" void kernel_launch(void* const* d_in, const int* in_sizes, int n_in,
                              void* d_out, int out_size, void* d_ws, size_t ws_size,
                              hipStream_t stream) {
    const float* outputs   = (const float*)d_in[0];
    const float* gt_boxes  = (const float*)d_in[1];
    const int*   gt_labels = (const int*)d_in[2];

    const int Bt    = in_sizes[0] / (SS * CELL_F);   // 16384
    const int cells = Bt * SS;                       // 802816 (multiple of 256)
    const int blocks = cells / TPB;                  // 3136

    float* partials = (float*)d_ws;                  // 3136 floats of scratch

    yolo_pass1<<<blocks, TPB, 0, stream>>>(outputs, gt_boxes, gt_labels, partials);
    yolo_pass2<<<1, TPB, 0, stream>>>(partials, (float*)d_out, blocks,
                                      1.0f / (float)Bt);
}